// Watson_30820685316370
// MI455X (gfx1250) — compile-verified
//
#include <hip/hip_runtime.h>
#include <math.h>

// Problem constants from the reference
#define NP 512      // P
#define NK 16       // K components
#define KSTEPS 128  // P / 4 (k-unroll of V_WMMA_F32_16X16X4_F32)
#define ROWS_PER_BLOCK 128
#define BFRAG_FLOATS (KSTEPS * 32 * 2)  // 8192 floats = 32 KB

typedef __attribute__((ext_vector_type(2))) float v2f;
typedef __attribute__((ext_vector_type(8))) float v8f;

// Workspace layout (floats):
//   [0..15]              offs[k]  = logC[k] + log_softmax(pi)[k]
//   [16..31]             kappa[k] copy
//   [32..32+8192)        B fragments (mu_unit in WMMA-B lane layout, inv_norm folded in)
//   [8224..8224+nblocks) per-block partial sums
#define WS_OFFS  0
#define WS_KAPPA 16
#define WS_BFRAG 32
#define WS_PART  (32 + BFRAG_FLOATS)

__global__ void watson_setup(const float* __restrict__ mu,
                             const float* __restrict__ kappa,
                             const float* __restrict__ pi,
                             float* __restrict__ ws) {
  __shared__ float s_inv[NK];
  __shared__ float s_kum[NK];
  const int tid = threadIdx.x;

  // mu column inverse norms (threads 16..31)
  if (tid >= 16 && tid < 32) {
    const int k = tid - 16;
    float ss = 0.f;
    for (int p = 0; p < NP; ++p) {
      float v = mu[p * NK + k];
      ss += v * v;
    }
    s_inv[k] = rsqrtf(ss);
  }

  // Kummer log-series M(0.5, 256, kappa) (threads 0..15), log-domain recurrence
  if (tid < NK) {
    const float kap = kappa[tid];
    const float lk = logf(kap);
    float logkum = 0.f, foo = 0.f;
    for (int j = 1; j <= 1000; ++j) {
      const float jf = (float)j;
      foo += logf((0.5f + jf - 1.0f) / (jf * (256.0f + jf - 1.0f))) + lk;
      // logaddexp(logkum, foo)
      const float mx = fmaxf(logkum, foo);
      const float mn = fminf(logkum, foo);
      logkum = mx + log1pf(expf(mn - mx));
    }
    s_kum[tid] = logkum;
  }
  __syncthreads();

  if (tid < NK) {
    // log_softmax over pi (redundant per-thread, trivial)
    float mx = -INFINITY;
    for (int i = 0; i < NK; ++i) mx = fmaxf(mx, pi[i]);
    float s = 0.f;
    for (int i = 0; i < NK; ++i) s += expf(pi[i] - mx);
    const float lsm = pi[tid] - (mx + logf(s));

    const float half_p = 256.0f;
    const float logSA = lgammaf(half_p) - logf(2.0f)
                        - half_p * logf(3.14159265358979323846f);
    ws[WS_OFFS + tid] = (logSA - s_kum[tid]) + lsm;
    ws[WS_KAPPA + tid] = kappa[tid];
  }
  __syncthreads();

  // Pack B fragments: for k-step s, lane l (n = l&15, hi = l>>4) holds
  // B[4s+2hi][n] and B[4s+2hi+1][n] in consecutive dwords (v_wmma B layout).
  // inv_norm folded in here so the GEMM directly produces mu_unit projections.
  for (int f = tid; f < BFRAG_FLOATS; f += blockDim.x) {
    const int c  = f & 1;
    const int l  = (f >> 1) & 31;
    const int s  = f >> 6;
    const int n  = l & 15;
    const int hi = l >> 4;
    const int kd = 4 * s + 2 * hi + c;  // P-dimension index
    ws[WS_BFRAG + f] = mu[kd * NK + n] * s_inv[n];
  }
}

__global__ void __launch_bounds__(256, 2)
watson_main(const float* __restrict__ X,
            const float* __restrict__ ws,
            float* __restrict__ partials,
            int nrows) {
  __shared__ float lds_b[BFRAG_FLOATS];
  __shared__ float s_offs[NK];
  __shared__ float s_kap[NK];
  __shared__ float s_wsum[8];

  const int tid = threadIdx.x;
  if (tid < NK) {
    s_offs[tid] = ws[WS_OFFS + tid];
    s_kap[tid]  = ws[WS_KAPPA + tid];
  }
  {
    const float* __restrict__ bf = ws + WS_BFRAG;
    for (int i = tid; i < BFRAG_FLOATS; i += 256) lds_b[i] = bf[i];
  }
  __syncthreads();

  const int lane = tid & 31;
  const int w    = tid >> 5;
  const int n    = lane & 15;  // component index for B/C roles
  const int hi   = lane >> 4;
  const int row0 = blockIdx.x * ROWS_PER_BLOCK + w * 16;

  // A-operand: this lane supplies row m = lane&15 of the 16x4 tile.
  int rowm = row0 + n;
  if (rowm > nrows - 1) rowm = nrows - 1;  // clamp: EXEC must stay all-ones for WMMA
  const float* __restrict__ xptr = X + (size_t)rowm * NP + hi * 2;
  const float* __restrict__ bptr = lds_b + lane * 2;

  v8f c = {0.f, 0.f, 0.f, 0.f, 0.f, 0.f, 0.f, 0.f};
#pragma unroll 4
  for (int s = 0; s < KSTEPS; ++s) {
    v2f a = *(const v2f*)(xptr + s * 4);        // global_load_b64: X[m][4s+2hi .. +1]
    v2f b = *(const v2f*)(bptr + s * 64);       // ds_load_b64: mu_unit fragment
    c = __builtin_amdgcn_wmma_f32_16x16x4_f32(
        /*neg_a=*/false, a, /*neg_b=*/false, b,
        /*c_mod=*/(short)0, c, /*reuse_a=*/false, /*reuse_b=*/false);
  }

  // Epilogue: C element c[v] holds proj for row (row0 + v + 8*hi), component n.
  const float offs = s_offs[n];
  const float kap  = s_kap[n];
  float local = 0.f;
#pragma unroll
  for (int v = 0; v < 8; ++v) {
    const float val = c[v];
    float t = offs + kap * val * val;
    // logsumexp across the 16 components = 16-lane-group shuffle reduction
    float mx = t;
    for (int o = 1; o < 16; o <<= 1) mx = fmaxf(mx, __shfl_xor(mx, o, 32));
    float e = expf(t - mx);
    for (int o = 1; o < 16; o <<= 1) e += __shfl_xor(e, o, 32);
    const float lse = mx + logf(e);
    const int row = row0 + v + hi * 8;
    if (n == 0 && row < nrows) local += lse;   // one contributor per 16-lane group
  }
  local += __shfl_xor(local, 16, 32);          // fold the two half-wave contributors
  if (lane == 0) s_wsum[w] = local;
  __syncthreads();
  if (tid == 0) {
    float bs = 0.f;
    for (int i = 0; i < 8; ++i) bs += s_wsum[i];
    partials[blockIdx.x] = bs;
  }
}

__global__ void watson_reduce(const float* __restrict__ partials, int nb,
                              float* __restrict__ out) {
  __shared__ float sh[256];
  float s = 0.f;
  for (int i = threadIdx.x; i < nb; i += 256) s += partials[i];
  sh[threadIdx.x] = s;
  __syncthreads();
  for (int st = 128; st > 0; st >>= 1) {
    if (threadIdx.x < st) sh[threadIdx.x] += sh[threadIdx.x + st];
    __syncthreads();
  }
  if (threadIdx.x == 0) out[0] = sh[0];
}

extern "C" void kernel_launch(void* const* d_in, const int* in_sizes, int n_in,
                              void* d_out, int out_size, void* d_ws, size_t ws_size,
                              hipStream_t stream) {
  const float* X     = (const float*)d_in[0];
  const float* mu    = (const float*)d_in[1];
  const float* kappa = (const float*)d_in[2];
  const float* pi    = (const float*)d_in[3];
  float* ws  = (float*)d_ws;
  float* out = (float*)d_out;

  const int nrows = in_sizes[0] / NP;                          // 200000
  const int nb    = (nrows + ROWS_PER_BLOCK - 1) / ROWS_PER_BLOCK;  // 1563

  watson_setup<<<1, 256, 0, stream>>>(mu, kappa, pi, ws);
  watson_main<<<nb, 256, 0, stream>>>(X, ws, ws + WS_PART, nrows);
  watson_reduce<<<1, 256, 0, stream>>>(ws + WS_PART, nb, out);
}